// MedianBlur_47399259079080
// MI455X (gfx1250) — compile-verified
//
#include <hip/hip_runtime.h>
#include <stdint.h>

// ---------------- configuration ----------------
#define TILE 16                 // 16x16 output pixels per workgroup (256 threads, 8 waves)
#define KW   9                  // median kernel size
#define PAD  4                  // KW/2
#define LW   (TILE + 2 * PAD)   // 24 : halo tile width (conflict-free for wave32 lane map)
#define LDS_ELEMS (LW * LW)     // 576 floats = 2.3 KB LDS

// ---------------- CDNA5 async global->LDS ----------------
#ifndef __has_builtin
#define __has_builtin(x) 0
#endif

#if __has_builtin(__builtin_amdgcn_global_load_async_to_lds_b32)
#define HAVE_ASYNC_LDS 1
#else
#define HAVE_ASYNC_LDS 0
#endif

// builtin signature (probed round 1): (__device__ int*, __shared__ int*, imm offset, imm cpol)
typedef __attribute__((address_space(1))) int as1_int;
typedef __attribute__((address_space(3))) int as3_int;

__device__ __forceinline__ void wait_async_lds() {
#if HAVE_ASYNC_LDS
#if __has_builtin(__builtin_amdgcn_s_wait_asynccnt)
    __builtin_amdgcn_s_wait_asynccnt(0);
#else
    asm volatile("s_wait_asynccnt 0" ::: "memory");
#endif
#endif
}

// compare-swap: a <- min, b <- max (v_min_num_f32 / v_max_num_f32)
__device__ __forceinline__ void cswap(float& a, float& b) {
    float lo = fminf(a, b);
    b = fmaxf(a, b);
    a = lo;
}

__global__ __launch_bounds__(256)
void median9_kernel(const float* __restrict__ in, float* __restrict__ out,
                    int H, int W) {
    __shared__ float tile[LDS_ELEMS];

    const int tid   = threadIdx.x;
    const int plane = blockIdx.z;
    const int by    = blockIdx.y * TILE;
    const int bx    = blockIdx.x * TILE;

    const float* planeIn = in + (size_t)plane * H * W;

    // ---- cooperative halo-tile fill with zero padding ----
    // In-bounds cells: async global->LDS load (ASYNCcnt).
    // Out-of-bounds cells: ds_store zero (DScnt).
    // Disjoint LDS addresses -> no ordering hazard; one s_wait_asynccnt + barrier
    // makes the tile visible to all 8 waves.
    for (int c = tid; c < LDS_ELEMS; c += 256) {
        const int r  = c / LW;
        const int cc = c - r * LW;
        const int gy = by + r  - PAD;
        const int gx = bx + cc - PAD;
        const bool inb = (gy >= 0) & (gy < H) & (gx >= 0) & (gx < W);
        if (inb) {
#if HAVE_ASYNC_LDS
            __builtin_amdgcn_global_load_async_to_lds_b32(
                (as1_int*)(uintptr_t)(planeIn + (size_t)gy * W + gx),
                (as3_int*)&tile[c],
                /*offset=*/0, /*cpol=*/0);
#else
            tile[c] = planeIn[(size_t)gy * W + gx];
#endif
        } else {
            tile[c] = 0.0f;
        }
    }
    wait_async_lds();
    __syncthreads();

    // ---- per-thread exact median of 81 via forgetful selection ----
    const int ty = tid >> 4;
    const int tx = tid & 15;
    const float* base = &tile[ty * LW + tx];   // window origin in LDS

    // buffer of ceil(81/2)+1 = 42 candidates, kept entirely in VGPRs
    float v[42];
#pragma unroll
    for (int j = 0; j < 42; ++j) {
        v[j] = base[(j / KW) * LW + (j % KW)];   // constant offsets after unroll
    }

    // 39 forget steps. Each step, over current buffer size s:
    //  1) pair split: cswap(v[i], v[i+h]), h = s/2  -> min in low half (+ v[s-1]
    //     if s odd, which stays a valid candidate for both extremes),
    //     max in high half
    //  2) sweep global min into v[0]   (low-half candidates)
    //  3) sweep global max into v[s-1] (high-half candidates)
    //  4) overwrite v[0] (discarded min) with the next window value,
    //     drop v[s-1] (discarded max) by shrinking s next iteration.
    // Both extremes are provably non-medians since the buffer stays >= 42-step
    // ~ ceil(81/2)+1 invariant. Cost ~3s/2-2 compare-swaps per step.
#pragma unroll
    for (int j = 42; j <= 80; ++j) {
        const int s = 84 - j;                    // current buffer size (42 .. 4)
        const int h = s / 2;
#pragma unroll
        for (int i = 0; i < h; ++i) cswap(v[i], v[i + h]);     // split
#pragma unroll
        for (int i = 1; i < h; ++i) cswap(v[0], v[i]);         // min -> v[0]
        if (s & 1) cswap(v[0], v[s - 1]);                      // odd leftover
#pragma unroll
        for (int i = h; i < s - 1; ++i) cswap(v[i], v[s - 1]); // max -> v[s-1]
        v[0] = base[(j / KW) * LW + (j % KW)];   // replace discarded min
    }

    // 3 survivors remain: median is their middle element (med3)
    const float a = v[0], b = v[1], c = v[2];
    const float mn = fminf(a, b);
    const float mx = fmaxf(a, b);
    const float med = fmaxf(mn, fminf(mx, c));

    out[(size_t)plane * H * W + (size_t)(by + ty) * W + (bx + tx)] = med;
}

extern "C" void kernel_launch(void* const* d_in, const int* in_sizes, int n_in,
                              void* d_out, int out_size, void* d_ws, size_t ws_size,
                              hipStream_t stream) {
    const float* img = (const float*)d_in[0];
    float* out = (float*)d_out;

    const int H = 512, W = 512;
    const int planes = in_sizes[0] / (H * W);    // 4 * 3 = 12

    dim3 grid(W / TILE, H / TILE, planes);       // (32, 32, 12)
    median9_kernel<<<grid, 256, 0, stream>>>(img, out, H, W);
}